// TemporalConvolution_8546984919605
// MI455X (gfx1250) — compile-verified
//
#include <hip/hip_runtime.h>
#include <hip/hip_bf16.h>

typedef __attribute__((ext_vector_type(2))) float v2f;
typedef __attribute__((ext_vector_type(8))) float v8f;

// out = X * mw^T, mw = weights masked to the {sub,super}-diagonal band.
// X: [rows x 128] fp32, rows = 64*25*256. One block = 16 rows x 128 outputs.
// 8 waves/block; wave w owns f-tile [16w, 16w+16). The band restricts useful
// K to [16w-1, 16w+16] -> fixed, fully-unrolled 6 steps of WMMA_F32_16X16X4_F32.
// Straight-line body: EXEC is never masked around the WMMAs.
__global__ __launch_bounds__(256) void banded_wmma_f32_kernel(
    const float* __restrict__ x,
    const float* __restrict__ w,
    float* __restrict__ out)
{
    constexpr int T = 128;
    const int lane = threadIdx.x & 31;
    const int wave = threadIdx.x >> 5;      // 0..7 -> which f tile
    const int half = lane >> 4;             // 0 or 1 (lane group)
    const int lm   = lane & 15;

    const long row0 = (long)blockIdx.x * 16;
    const int  f    = wave * 16 + lm;       // output column for B/D fragments

    const float* xrow = x + (row0 + lm) * T;   // A fragment source row

    // Only two weights ever contribute to column f: w[f,f-1] and w[f,f+1].
    // Clamp the load addresses (branch-free, in-bounds); the selects below can
    // never pick the clamped value at the boundaries because f-1 = -1 and
    // f+1 = 128 match no k in [0,128).
    const int fm1 = f - 1;                  // may be -1 (never matches any k)
    const int fp1 = f + 1;                  // may be 128 (never matches any k)
    const float wsub = w[max(f * T + f - 1, 0)];
    const float wsup = w[min(f * T + f + 1, T * T - 1)];

    v8f acc = {0.f, 0.f, 0.f, 0.f, 0.f, 0.f, 0.f, 0.f};

    // Fixed 6 K-blocks covering the band [16w-4, 16w+20); out-of-range blocks
    // (only at wave 0 / wave 7 edges) get B == 0 and contribute nothing.
    #pragma unroll
    for (int i = 0; i < 6; ++i) {
        const int kb    = 4 * wave - 1 + i;
        const int kbc   = min(max(kb, 0), 31);       // clamped (safe) K-block
        const float vld = (kb == kbc) ? 1.0f : 0.0f; // 0 for the clamped dups
        const int k0    = kbc * 4 + half * 2;        // this lane's K pair

        // A fragment (16x4 fp32): lane -> M = lm, K = k0 + {0,1}. 8B aligned.
        v2f a = *(const v2f*)(xrow + k0);

        // B fragment (4x16 fp32): lane -> N = lm (col f), K = k0 + {0,1}.
        float bx = (k0     == fm1) ? wsub : ((k0     == fp1) ? wsup : 0.0f);
        float by = (k0 + 1 == fm1) ? wsub : ((k0 + 1 == fp1) ? wsup : 0.0f);
        v2f b;
        b.x = bx * vld;
        b.y = by * vld;

        // D = A*B + C  (fp32 WMMA; accumulator chained through C)
        acc = __builtin_amdgcn_wmma_f32_16x16x4_f32(
            /*neg_a=*/false, a, /*neg_b=*/false, b,
            /*c_mod=*/(short)0, acc, /*reuse_a=*/false, /*reuse_b=*/false);
    }

    // D fragment (16x16 fp32, 8 VGPRs): VGPR v -> row M = v + 8*half, col N = lm.
    // Output is written once and never re-read by this grid -> non-temporal.
    float* obase = out + (row0 + 8 * half) * T + wave * 16 + lm;
    #pragma unroll
    for (int v = 0; v < 8; ++v) {
        __builtin_nontemporal_store(acc[v], obase + (long)v * T);
    }
}

// Scalar 2-tap stencil tail for any leftover rows (rows % 16). With the given
// shapes rows = 409600 is divisible by 16, so this normally launches 0 blocks.
__global__ void banded_tail_kernel(
    const float* __restrict__ x,
    const float* __restrict__ w,
    float* __restrict__ out,
    long row_start, long n_rows)
{
    constexpr int T = 128;
    long idx = (long)blockIdx.x * blockDim.x + threadIdx.x;
    long total = n_rows * T;
    if (idx >= total) return;
    long r = row_start + idx / T;
    int  fcol = (int)(idx % T);
    const float* xr = x + r * T;
    float acc = 0.0f;
    if (fcol >= 1)     acc += w[(long)fcol * T + (fcol - 1)] * xr[fcol - 1];
    if (fcol <= T - 2) acc += w[(long)fcol * T + (fcol + 1)] * xr[fcol + 1];
    out[r * T + fcol] = acc;
}

extern "C" void kernel_launch(void* const* d_in, const int* in_sizes, int n_in,
                              void* d_out, int out_size, void* d_ws, size_t ws_size,
                              hipStream_t stream) {
    (void)n_in; (void)d_ws; (void)ws_size; (void)out_size;
    constexpr int T = 128;
    const float* x = (const float*)d_in[0];   // [64,25,256,128] fp32
    const float* w = (const float*)d_in[1];   // [128,128] fp32
    float* out = (float*)d_out;

    const long rows = (long)in_sizes[0] / T;  // 409600
    const long full = rows / 16;              // WMMA row-blocks
    const long rem  = rows - full * 16;

    if (full > 0) {
        banded_wmma_f32_kernel<<<dim3((unsigned)full), dim3(256), 0, stream>>>(x, w, out);
    }
    if (rem > 0) {
        long tail_elems = rem * T;
        int  blk = 256;
        long grd = (tail_elems + blk - 1) / blk;
        banded_tail_kernel<<<dim3((unsigned)grd), dim3(blk), 0, stream>>>(
            x, w, out, full * 16, rem);
    }
}